// RobustRFSQBlock_22686017258069
// MI455X (gfx1250) — compile-verified
//
#include <hip/hip_runtime.h>
#include <math.h>

#ifndef __has_builtin
#define __has_builtin(x) 0
#endif

#define AS1 __attribute__((address_space(1)))
#define AS3 __attribute__((address_space(3)))

typedef float v4f __attribute__((ext_vector_type(4)));
typedef int   v4i __attribute__((vector_size(16)));   // matches builtin's printed param type

#define DIMS     1024
#define LAYERS   8
#define WAVES    8            // rows per block (1 wave32 per row)
#define TPB      (WAVES * 32)
#define CHUNKS   8            // float4 chunks per lane: 8*4 = 32 elements/lane
#define EPL      (CHUNKS * 4) // elements per lane

#if __has_builtin(__builtin_amdgcn_global_load_async_to_lds_b128)
#define HAVE_ASYNC_LDS 1
#else
#define HAVE_ASYNC_LDS 0
#endif

__device__ __forceinline__ void wait_asynccnt0() {
#if __has_builtin(__builtin_amdgcn_s_wait_asynccnt)
  __builtin_amdgcn_s_wait_asynccnt(0);
#else
  asm volatile("s_wait_asynccnt 0" ::: "memory");
#endif
}

__global__ __launch_bounds__(TPB) void rfsq_kernel(
    const float* __restrict__ z, const float* __restrict__ bnd,
    float* __restrict__ out_q, float* __restrict__ out_codes, int rows) {
  __shared__ float lds_z[WAVES * DIMS];  // 32 KB: original rows, kept for qsum epilogue

  const int lane = threadIdx.x & 31;
  const int wave = threadIdx.x >> 5;
  const int row  = blockIdx.x * WAVES + wave;
  if (row >= rows) return;  // wave-uniform; no barriers used anywhere

  const float* zr = z + (size_t)row * DIMS;
  float* lz = &lds_z[wave * DIMS];

  // ---- stage row into LDS (CDNA5 async global->LDS path, ASYNCcnt) ----
#if HAVE_ASYNC_LDS
#pragma unroll
  for (int c = 0; c < CHUNKS; ++c) {
    const int off = c * 128 + lane * 4;
    __builtin_amdgcn_global_load_async_to_lds_b128(
        (AS1 v4i*)(zr + off), (AS3 v4i*)(lz + off), 0, 0);
  }
  wait_asynccnt0();
  asm volatile("" ::: "memory");  // keep ds_loads below the wait
#else
#pragma unroll
  for (int c = 0; c < CHUNKS; ++c) {
    const int off = c * 128 + lane * 4;
    *(v4f*)(lz + off) = *(const v4f*)(zr + off);
  }
#endif

  // boundaries = linspace(-1,1,7): reconstruct affine map from the buffer
  const float b0 = bnd[0];
  const float b6 = bnd[6];
  const float step  = (b6 - b0) * (1.0f / 6.0f);
  const float istep = 1.0f / step;

  // ---- residual into registers (ds_load_b128) ----
  float r[EPL];
#pragma unroll
  for (int c = 0; c < CHUNKS; ++c) {
    v4f v = *(v4f*)(lz + c * 128 + lane * 4);
    r[c * 4 + 0] = v.x; r[c * 4 + 1] = v.y;
    r[c * 4 + 2] = v.z; r[c * 4 + 3] = v.w;
  }

  unsigned cpack[EPL];
#pragma unroll
  for (int e = 0; e < EPL; ++e) cpack[e] = 0u;

  // ---- 8 residual-quantization layers, fully register/LDS resident ----
#pragma unroll 1
  for (int l = 0; l < LAYERS; ++l) {
    float psum = 0.0f, pss = 0.0f;
#pragma unroll
    for (int e = 0; e < EPL; ++e) {
      psum += r[e];
      pss = fmaf(r[e], r[e], pss);
    }
    // wave32 butterfly reduction (cross-lane, no LDS memory, no barrier)
#pragma unroll
    for (int m = 16; m >= 1; m >>= 1) {
      psum += __shfl_xor(psum, m, 32);
      pss  += __shfl_xor(pss,  m, 32);
    }
    const float mean = psum * (1.0f / DIMS);
    float var = (pss - psum * mean) * (1.0f / (DIMS - 1));
    var = fmaxf(var, 0.0f);
    const float s     = sqrtf(var) + 1e-5f;  // std (ddof=1) + EPS
    const float inv_s = 1.0f / s;

#pragma unroll
    for (int e = 0; e < EPL; ++e) {
      const float zn = (r[e] - mean) * inv_s;
      float f = rintf((zn - b0) * istep);       // nearest of 7 uniform levels
      f = fminf(fmaxf(f, 0.0f), 6.0f);          // == argmin |zn - boundaries|
      cpack[e] |= ((unsigned)(int)f) << (l * 4);
      const float bval = fmaf(f, step, b0);     // boundaries[idx]
      const float zq   = fmaf(bval, s, mean);   // de-normalize
      r[e] -= zq;                               // residual update
    }
  }

  // ---- epilogue: qsum = z - residual_final (z re-read from LDS) ----
  float* oq = out_q + (size_t)row * DIMS;
#pragma unroll
  for (int c = 0; c < CHUNKS; ++c) {
    const int off = c * 128 + lane * 4;
    v4f zv = *(v4f*)(lz + off);
    v4f qv;
    qv.x = zv.x - r[c * 4 + 0];
    qv.y = zv.y - r[c * 4 + 1];
    qv.z = zv.z - r[c * 4 + 2];
    qv.w = zv.w - r[c * 4 + 3];
    *(v4f*)(oq + off) = qv;
  }

  // ---- codes: [row, d, 8] as floats; 2x b128 stores per element ----
  float* oc = out_codes + (size_t)row * DIMS * LAYERS;
#pragma unroll
  for (int e = 0; e < EPL; ++e) {
    const int d = (e >> 2) * 128 + lane * 4 + (e & 3);
    const unsigned p = cpack[e];
    v4f c0, c1;
    c0.x = (float)( p         & 15u);
    c0.y = (float)((p >>  4)  & 15u);
    c0.z = (float)((p >>  8)  & 15u);
    c0.w = (float)((p >> 12)  & 15u);
    c1.x = (float)((p >> 16)  & 15u);
    c1.y = (float)((p >> 20)  & 15u);
    c1.z = (float)((p >> 24)  & 15u);
    c1.w = (float)( p >> 28        );
    *(v4f*)(oc + (size_t)d * LAYERS)     = c0;
    *(v4f*)(oc + (size_t)d * LAYERS + 4) = c1;
  }
}

extern "C" void kernel_launch(void* const* d_in, const int* in_sizes, int n_in,
                              void* d_out, int out_size, void* d_ws, size_t ws_size,
                              hipStream_t stream) {
  const float* z   = (const float*)d_in[0];
  const float* bnd = (const float*)d_in[1];
  const int total = in_sizes[0];       // B*S*D = 8388608
  const int rows  = total / DIMS;      // 8192
  float* out_q     = (float*)d_out;
  float* out_codes = out_q + (size_t)total;
  const int blocks = (rows + WAVES - 1) / WAVES;
  rfsq_kernel<<<blocks, TPB, 0, stream>>>(z, bnd, out_q, out_codes, rows);
}